// SimpleNetwork_58239756534442
// MI455X (gfx1250) — compile-verified
//
#include <hip/hip_runtime.h>
#include <hip/hip_bf16.h>

typedef __attribute__((ext_vector_type(2))) float v2f;
typedef __attribute__((ext_vector_type(8))) float v8f;

#define FDIM 128

// ---------------------------------------------------------------------------
// Step 0: re-initialize workspace every call (ws is poisoned / not restored).
// deg[v] = 1.0 (self-loop), s[v] = 0, y[k] = 0.
// ---------------------------------------------------------------------------
__global__ void gcn_init_kernel(float* __restrict__ deg, float* __restrict__ s,
                                float* __restrict__ y, int n) {
  int i = blockIdx.x * blockDim.x + threadIdx.x;
  if (i < n) {
    deg[i] = 1.0f;   // self-loop contributes 1 to every node's degree
    s[i]   = 0.0f;
  }
  if (i < FDIM) y[i] = 0.0f;
}

// ---------------------------------------------------------------------------
// Step 1: deg[dst] += 1 per edge.
// ---------------------------------------------------------------------------
__global__ void gcn_deg_kernel(const long long* __restrict__ dst,
                               float* __restrict__ deg, int e) {
  int i = blockIdx.x * blockDim.x + threadIdx.x;
  if (i < e) atomicAdd(&deg[(int)dst[i]], 1.0f);
}

// ---------------------------------------------------------------------------
// Step 2: dinv[v] = deg[v]^-0.5 (deg >= 1 always due to self-loops).
// ---------------------------------------------------------------------------
__global__ void gcn_dinv_kernel(float* __restrict__ deg, int n) {
  int i = blockIdx.x * blockDim.x + threadIdx.x;
  if (i < n) deg[i] = rsqrtf(deg[i]);
}

// ---------------------------------------------------------------------------
// Step 3: s[src] += dinv[dst] per edge.
// ---------------------------------------------------------------------------
__global__ void gcn_srcsum_kernel(const long long* __restrict__ src,
                                  const long long* __restrict__ dst,
                                  const float* __restrict__ dinv,
                                  float* __restrict__ s, int e) {
  int i = blockIdx.x * blockDim.x + threadIdx.x;
  if (i < e) atomicAdd(&s[(int)src[i]], dinv[(int)dst[i]]);
}

// ---------------------------------------------------------------------------
// Step 4: c[v] = dinv[v]*(s[v] + dinv[v])   (edge terms + self-loop dinv^2)
//         stored back in-place into s.
// ---------------------------------------------------------------------------
__global__ void gcn_coef_kernel(const float* __restrict__ dinv,
                                float* __restrict__ s, int n) {
  int i = blockIdx.x * blockDim.x + threadIdx.x;
  if (i < n) {
    float d = dinv[i];
    s[i] = d * (s[i] + d);
  }
}

// ---------------------------------------------------------------------------
// Step 5: y[t] = sum_v c[v] * x[v][t]   (the 51 MB streaming pass)
// blockDim.x == FDIM; thread t owns feature column t; block-strided over nodes.
// ---------------------------------------------------------------------------
__global__ void gcn_colsum_kernel(const float* __restrict__ x,
                                  const float* __restrict__ c,
                                  float* __restrict__ y, int n) {
  int t = threadIdx.x;           // 0..127
  int stride = gridDim.x;
  float acc = 0.0f;
  for (int v = blockIdx.x; v < n; v += stride) {
    int vn = v + stride;
    if (vn < n) {
      // CDNA5 speculative prefetch of the next row this block will touch
      __builtin_prefetch(x + (size_t)vn * FDIM + t, 0, 1);
    }
    acc = fmaf(c[v], x[(size_t)v * FDIM + t], acc);
  }
  atomicAdd(&y[t], acc);
}

// ---------------------------------------------------------------------------
// Step 6: out[f] = sum_k y[k] * W[f][k] + n * bias[f]
// via V_WMMA_F32_16X16X4_F32.  One block, 8 waves; wave w computes output
// features [16w, 16w+16).  A has y in row M=0 only; B[k][n] = W[nbase+n][k].
//
// 32-bit A 16x4 layout (ISA 7.12.2): lanes 0-15 -> M=lane, VGPR0=K0 VGPR1=K1;
// lanes 16-31 -> M=lane-16, VGPR0=K2 VGPR1=K3.  B 4x16 mirrors the K split
// across lane halves with N = lane%16.  D row 0 = acc[0] of lanes 0-15.
//
// y is loaded UNCONDITIONALLY by every lane (broadcast address -> L0 hit)
// and masked into row 0 with selects, so no exec-divergence surrounds the
// WMMA issue stream; EXEC stays all-ones throughout the loop.
// ---------------------------------------------------------------------------
__global__ void gcn_matvec_wmma_kernel(const float* __restrict__ W,
                                       const float* __restrict__ bias,
                                       const float* __restrict__ y,
                                       float* __restrict__ out, int nnodes) {
  const int lane = threadIdx.x & 31;
  const int wave = threadIdx.x >> 5;   // 0..7 -> output tile
  const int half = lane >> 4;          // K-half selector
  const int col  = lane & 15;          // M (for A) / N (for B)
  const int nbase = wave * 16;
  const int f = nbase + col;           // output feature this lane's B column
  const float amask = (col == 0) ? 1.0f : 0.0f;   // select, not branch

  const float* __restrict__ wrow = W + (size_t)f * FDIM;

  v8f acc = {};
#pragma unroll
  for (int kb = 0; kb < FDIM; kb += 4) {
    const int k0 = kb + 2 * half;      // K index for vector element 0
    // All lanes load y (uniform per half-wave); mask picks row M==0 only.
    const float ya = y[k0 + 0];
    const float yb = y[k0 + 1];
    v2f a, b;
    a.x = ya * amask;
    a.y = yb * amask;
    // B[k][n] = W^T[k][f] = W[f][k]  (W row-major [f,k])
    b.x = wrow[k0 + 0];
    b.y = wrow[k0 + 1];
    acc = __builtin_amdgcn_wmma_f32_16x16x4_f32(
        /*neg_a=*/false, a, /*neg_b=*/false, b,
        /*c_mod=*/(short)0, acc, /*reuse_a=*/false, /*reuse_b=*/false);
  }
  // D[M=0][N=col] lives in acc[0] of lanes 0-15 (half==0).
  if (half == 0) {
    out[f] = acc[0] + (float)nnodes * bias[f];
  }
}

// ---------------------------------------------------------------------------
extern "C" void kernel_launch(void* const* d_in, const int* in_sizes, int n_in,
                              void* d_out, int out_size, void* d_ws, size_t ws_size,
                              hipStream_t stream) {
  const float*     x    = (const float*)d_in[0];       // [n, 128]
  const float*     W    = (const float*)d_in[1];       // [128, 128]
  const float*     bias = (const float*)d_in[2];       // [128]
  const long long* ei   = (const long long*)d_in[3];   // [2, e] int64

  const int n = in_sizes[0] / FDIM;
  const int e = in_sizes[3] / 2;
  const long long* src = ei;
  const long long* dst = ei + e;

  // Workspace layout: deg/dinv [n] | s/c [n] | y [128]
  float* deg = (float*)d_ws;
  float* s   = deg + n;
  float* y   = s + n;

  const int nb_n = (n + 255) / 256;
  const int nb_e = (e + 255) / 256;

  gcn_init_kernel<<<nb_n, 256, 0, stream>>>(deg, s, y, n);
  gcn_deg_kernel<<<nb_e, 256, 0, stream>>>(dst, deg, e);
  gcn_dinv_kernel<<<nb_n, 256, 0, stream>>>(deg, n);
  gcn_srcsum_kernel<<<nb_e, 256, 0, stream>>>(src, dst, deg, s, e);
  gcn_coef_kernel<<<nb_n, 256, 0, stream>>>(deg, s, n);

  int cs_blocks = 1024;
  if (cs_blocks > n) cs_blocks = n;
  gcn_colsum_kernel<<<cs_blocks, FDIM, 0, stream>>>(x, s, y, n);

  gcn_matvec_wmma_kernel<<<1, 256, 0, stream>>>(W, bias, y, (float*)d_out, n);
}